// TensorGraphSpatial2DConvolution_35493609734613
// MI455X (gfx1250) — compile-verified
//
#include <hip/hip_runtime.h>
#include <math.h>

// ---------------- problem constants ----------------
#define BB   8
#define CC   128
#define HH   48
#define WW_  48
#define HW   2304          // 48*48
#define RR   4
#define DD   32
#define CHW  294912        // 128*2304
#define HWHW 5308416       // 2304*2304
#define XN   256           // B*d
#define XSZ  589824        // 2304*256
#define PADP 2500          // 50*50
#define PADC 320000        // 128*2500

typedef float v2f __attribute__((ext_vector_type(2)));
typedef float v8f __attribute__((ext_vector_type(8)));

__device__ __forceinline__ v8f wmma4(v2f a, v2f b, v8f c) {
  // D = A(16x4) * B(4x16) + C, fp32 WMMA (CDNA5)
  return __builtin_amdgcn_wmma_f32_16x16x4_f32(false, a, false, b, (short)0, c,
                                               false, false);
}

// FFT butterfly coefficients for A-components: A0, A2, Ar, Ai
__constant__ float c_bfly[4][4] = {
    {1.f, 1.f, 1.f, 1.f},    // A0 = a0+a1+a2+a3
    {1.f, -1.f, 1.f, -1.f},  // A2 = a0-a1+a2-a3
    {1.f, 0.f, -1.f, 0.f},   // Ar = a0-a2
    {0.f, -1.f, 0.f, 1.f}};  // Ai = a3-a1
// 6 products: A0X0, A2X2, ArXr, AiXi, ArXi, AiXr
__constant__ int c_pa[6] = {0, 1, 2, 3, 2, 3};
__constant__ int c_px[6] = {0, 1, 2, 3, 3, 2};

// ---------------- 1x1 conv: per-batch  Y = W(128x128) @ X(128x2304) + b ----
// block = 4 waves; wave tile = 32(M) x 64(N); block covers all 128 M rows.
__global__ __launch_bounds__(128) void k_conv1x1(const float* __restrict__ X,
                                                 const float* __restrict__ W,
                                                 const float* __restrict__ bias,
                                                 float* __restrict__ Y) {
  const int b = blockIdx.z;
  const int lane = threadIdx.x & 31, wave = threadIdx.x >> 5;
  const int m0 = wave * 32;
  const int n0 = blockIdx.x * 64;
  const int h = lane >> 4, l16 = lane & 15;
  const float* xb = X + b * CHW;
  v8f acc[2][4] = {};
  for (int k0 = 0; k0 < 128; k0 += 4) {
    const int kk = k0 + 2 * h;
    v2f af[2];
#pragma unroll
    for (int i = 0; i < 2; i++)
      af[i] = *(const v2f*)(W + (m0 + i * 16 + l16) * 128 + kk);
#pragma unroll
    for (int t = 0; t < 4; t++) {
      const int n = n0 + t * 16 + l16;
      v2f bf;
      bf.x = xb[kk * HW + n];
      bf.y = xb[(kk + 1) * HW + n];
#pragma unroll
      for (int i = 0; i < 2; i++) acc[i][t] = wmma4(af[i], bf, acc[i][t]);
    }
  }
  float* yb = Y + b * CHW;
#pragma unroll
  for (int i = 0; i < 2; i++) {
#pragma unroll
    for (int t = 0; t < 4; t++) {
      const int n = n0 + t * 16 + l16;
#pragma unroll
      for (int p = 0; p < 8; p++) {
        const int row = m0 + i * 16 + p + 8 * h;
        yb[row * HW + n] = acc[i][t][p] + bias[row];
      }
    }
  }
}

// ---------------- GroupNorm(1) stats: per-sample sum / sumsq ---------------
__global__ __launch_bounds__(256) void k_gnstats(const float* __restrict__ t,
                                                 float* __restrict__ stats) {
  const int b = blockIdx.y;
  const float* p = t + b * CHW + blockIdx.x * 4096;
  float s = 0.f, ss = 0.f;
  for (int i = threadIdx.x; i < 4096; i += 256) {
    const float v = p[i];
    s += v;
    ss += v * v;
  }
  __shared__ float sh1[256], sh2[256];
  sh1[threadIdx.x] = s;
  sh2[threadIdx.x] = ss;
  __syncthreads();
  for (int o = 128; o > 0; o >>= 1) {
    if (threadIdx.x < o) {
      sh1[threadIdx.x] += sh1[threadIdx.x + o];
      sh2[threadIdx.x] += sh2[threadIdx.x + o];
    }
    __syncthreads();
  }
  if (threadIdx.x == 0) {
    atomicAdd(&stats[b * 2 + 0], sh1[0]);
    atomicAdd(&stats[b * 2 + 1], sh2[0]);
  }
}

// ---------------- GroupNorm apply + ReLU (in place) ------------------------
__global__ __launch_bounds__(256) void k_gnapply(float* __restrict__ t,
                                                 const float* __restrict__ stats,
                                                 const float* __restrict__ g,
                                                 const float* __restrict__ be) {
  const int idx = blockIdx.x * 256 + threadIdx.x;  // < 2359296
  const int b = idx / CHW;
  const int c = (idx % CHW) / HW;
  const float inv = 1.f / (float)CHW;
  const float mu = stats[b * 2] * inv;
  const float var = stats[b * 2 + 1] * inv - mu * mu;
  const float r = rsqrtf(var + 1e-5f);
  const float v = (t[idx] - mu) * r * g[c] + be[c];
  t[idx] = fmaxf(v, 0.f);
}

// ---------------- sigmoid gate * x, + per-(b,r) sums for SE ----------------
__global__ __launch_bounds__(256) void k_sigmul(float* __restrict__ t,
                                                const float* __restrict__ x,
                                                float* __restrict__ sesum) {
  const int ch = blockIdx.x;  // 0..1023 = b*128+c
  const int b = ch >> 7, c = ch & 127, r = c >> 5;
  float* tp = t + ch * HW;
  const float* xp = x + ch * HW;
  float s = 0.f;
  for (int i = threadIdx.x; i < HW; i += 256) {
    const float a = 1.f / (1.f + expf(-tp[i]));
    const float v = a * xp[i];
    tp[i] = v;
    s += v;
  }
  __shared__ float sh[256];
  sh[threadIdx.x] = s;
  __syncthreads();
  for (int o = 128; o > 0; o >>= 1) {
    if (threadIdx.x < o) sh[threadIdx.x] += sh[threadIdx.x + o];
    __syncthreads();
  }
  if (threadIdx.x == 0) atomicAdd(&sesum[b * 4 + r], sh[0]);
}

// ---------------- tiny SE block: f2[b,r] -----------------------------------
__global__ void k_se(const float* __restrict__ sesum,
                     const float* __restrict__ w1, const float* __restrict__ b1,
                     const float* __restrict__ w2, const float* __restrict__ b2,
                     float* __restrict__ f2) {
  const int t = threadIdx.x;
  if (t >= 32) return;
  const int b = t >> 2, r = t & 3;
  float sq[4];
#pragma unroll
  for (int i = 0; i < 4; i++) sq[i] = sesum[b * 4 + i] * (1.f / 73728.f);
  float f1[2];
#pragma unroll
  for (int j = 0; j < 2; j++) {
    float a = b1[j];
#pragma unroll
    for (int i = 0; i < 4; i++) a += sq[i] * w1[j * 4 + i];
    f1[j] = fmaxf(a, 0.f);
  }
  float a = b2[r];
#pragma unroll
  for (int j = 0; j < 2; j++) a += f1[j] * w2[r * 2 + j];
  f2[t] = 1.f / (1.f + expf(-a));
}

// ---------------- X butterflies (scaled by f2): X0,X2,Xr,Xi (2304x256) -----
__global__ __launch_bounds__(256) void k_bflyx(const float* __restrict__ xa,
                                               const float* __restrict__ f2,
                                               float* __restrict__ Xc) {
  const int idx = blockIdx.x * 256 + threadIdx.x;  // < 589824
  const int m = idx >> 8, col = idx & 255, b = col >> 5, dd = col & 31;
  const float* base = xa + b * CHW + m * DD + dd;
  const float v0 = f2[b * 4 + 0] * base[0];
  const float v1 = f2[b * 4 + 1] * base[73728];
  const float v2 = f2[b * 4 + 2] * base[147456];
  const float v3 = f2[b * 4 + 3] * base[221184];
  Xc[idx] = v0 + v1 + v2 + v3;
  Xc[XSZ + idx] = v0 - v1 + v2 - v3;
  Xc[2 * XSZ + idx] = v0 - v2;
  Xc[3 * XSZ + idx] = v3 - v1;
}

// ---------------- big GEMM: P[prod] = comp(adj) @ Xc[xc] -------------------
// block = 4 waves; wave tile = 32(M) x 64(N); block covers 128(M) x 64(N).
__global__ __launch_bounds__(128) void k_bigmm(const float* __restrict__ adj,
                                               const float* __restrict__ Xc,
                                               float* __restrict__ P) {
  const int prod = blockIdx.z;
  const int ac = c_pa[prod], xc = c_px[prod];
  const int lane = threadIdx.x & 31, wave = threadIdx.x >> 5;
  const int m0 = blockIdx.y * 128 + wave * 32;
  const int n0 = blockIdx.x * 64;
  const int h = lane >> 4, l16 = lane & 15;
  const float cf0 = c_bfly[ac][0], cf1 = c_bfly[ac][1];
  const float cf2 = c_bfly[ac][2], cf3 = c_bfly[ac][3];
  const float* Xb = Xc + xc * XSZ;
  const float* arow0 = adj + (m0 + l16) * HW;
  const float* arow1 = adj + (m0 + 16 + l16) * HW;
  v8f acc[2][4] = {};
  for (int k0 = 0; k0 < HW; k0 += 4) {
    const int kk = k0 + 2 * h;
    v2f af[2];
#pragma unroll
    for (int i = 0; i < 2; i++) {
      const float* arow = (i == 0) ? arow0 : arow1;
      v2f a = {0.f, 0.f};
      if (cf0 != 0.f) { v2f s = *(const v2f*)(arow + kk);            a.x += cf0 * s.x; a.y += cf0 * s.y; }
      if (cf1 != 0.f) { v2f s = *(const v2f*)(arow + HWHW + kk);     a.x += cf1 * s.x; a.y += cf1 * s.y; }
      if (cf2 != 0.f) { v2f s = *(const v2f*)(arow + 2 * HWHW + kk); a.x += cf2 * s.x; a.y += cf2 * s.y; }
      if (cf3 != 0.f) { v2f s = *(const v2f*)(arow + 3 * HWHW + kk); a.x += cf3 * s.x; a.y += cf3 * s.y; }
      af[i] = a;
    }
#pragma unroll
    for (int t = 0; t < 4; t++) {
      const int n = n0 + t * 16 + l16;
      v2f bf;
      bf.x = Xb[kk * XN + n];
      bf.y = Xb[(kk + 1) * XN + n];
#pragma unroll
      for (int i = 0; i < 2; i++) acc[i][t] = wmma4(af[i], bf, acc[i][t]);
    }
  }
  float* pp = P + prod * XSZ;
#pragma unroll
  for (int i = 0; i < 2; i++) {
#pragma unroll
    for (int t = 0; t < 4; t++) {
      const int n = n0 + t * 16 + l16;
#pragma unroll
      for (int p = 0; p < 8; p++) {
        pp[(m0 + i * 16 + p + 8 * h) * XN + n] = acc[i][t][p];
      }
    }
  }
}

// ---------------- iFFT recombination -> gc image (B,128,48,48) -------------
__global__ __launch_bounds__(256) void k_combine(const float* __restrict__ P,
                                                 float* __restrict__ gc) {
  const int idx = blockIdx.x * 256 + threadIdx.x;  // < 589824
  const int m = idx >> 8, col = idx & 255, b = col >> 5, dd = col & 31;
  const float y0 = P[idx];
  const float y2 = P[XSZ + idx];
  const float yr = P[2 * XSZ + idx] - P[3 * XSZ + idx];
  const float yi = P[4 * XSZ + idx] + P[5 * XSZ + idx];
  float* g = gc + b * CHW + m * DD + dd;
  g[0] = 0.25f * (y0 + y2 + 2.f * yr);
  g[73728] = 0.25f * (y0 - y2 - 2.f * yi);
  g[147456] = 0.25f * (y0 + y2 - 2.f * yr);
  g[221184] = 0.25f * (y0 - y2 + 2.f * yi);
}

// ---------------- zero-pad gc -> (B,128,50,50) -----------------------------
__global__ __launch_bounds__(256) void k_pad(const float* __restrict__ gc,
                                             float* __restrict__ pad) {
  const int idx = blockIdx.x * 256 + threadIdx.x;  // == 2560000 total
  const int b = idx / PADC;
  const int rem = idx % PADC;
  const int c = rem / PADP;
  const int p = rem % PADP;
  const int py = p / 50, px = p % 50;
  float v = 0.f;
  if (py >= 1 && py <= 48 && px >= 1 && px <= 48)
    v = gc[b * CHW + c * HW + (py - 1) * 48 + (px - 1)];
  pad[idx] = v;
}

// ---------------- 3x3 conv as implicit GEMM over padded input --------------
// block = 4 waves; wave tile = 32(M) x 64(N); block covers all 128 M rows.
__global__ __launch_bounds__(128) void k_conv3(const float* __restrict__ pad,
                                               const float* __restrict__ w,
                                               float* __restrict__ out) {
  const int b = blockIdx.z;
  const int lane = threadIdx.x & 31, wave = threadIdx.x >> 5;
  const int m0 = wave * 32;
  const int n0 = blockIdx.x * 64;
  const int h = lane >> 4, l16 = lane & 15;
  const float* pb = pad + b * PADC;
  int py[4], px[4];
#pragma unroll
  for (int t = 0; t < 4; t++) {
    const int n = n0 + t * 16 + l16;
    py[t] = n / 48;
    px[t] = n % 48;
  }
  v8f acc[2][4] = {};
  for (int ky = 0; ky < 3; ky++) {
    for (int kx = 0; kx < 3; kx++) {
      for (int k0 = 0; k0 < 128; k0 += 4) {
        const int kk = k0 + 2 * h;
        v2f af[2];
#pragma unroll
        for (int i = 0; i < 2; i++) {
          const float* wb =
              w + ((m0 + i * 16 + l16) * 128 + kk) * 9 + ky * 3 + kx;
          af[i].x = wb[0];
          af[i].y = wb[9];
        }
#pragma unroll
        for (int t = 0; t < 4; t++) {
          const int pin = (py[t] + ky) * 50 + px[t] + kx;
          v2f bf;
          bf.x = pb[kk * PADP + pin];
          bf.y = pb[(kk + 1) * PADP + pin];
#pragma unroll
          for (int i = 0; i < 2; i++) acc[i][t] = wmma4(af[i], bf, acc[i][t]);
        }
      }
    }
  }
  float* ob = out + b * CHW;
#pragma unroll
  for (int i = 0; i < 2; i++) {
#pragma unroll
    for (int t = 0; t < 4; t++) {
      const int n = n0 + t * 16 + l16;
#pragma unroll
      for (int p = 0; p < 8; p++) {
        ob[(m0 + i * 16 + p + 8 * h) * HW + n] = acc[i][t][p];
      }
    }
  }
}

// ---------------- launcher -------------------------------------------------
extern "C" void kernel_launch(void* const* d_in, const int* in_sizes, int n_in,
                              void* d_out, int out_size, void* d_ws,
                              size_t ws_size, hipStream_t stream) {
  const float* x = (const float*)d_in[0];
  const float* adj = (const float*)d_in[1];
  const float* sa_w1 = (const float*)d_in[2];
  const float* sa_b1 = (const float*)d_in[3];
  const float* gn_g1 = (const float*)d_in[4];
  const float* gn_be1 = (const float*)d_in[5];
  const float* sa_w2 = (const float*)d_in[6];
  const float* sa_b2 = (const float*)d_in[7];
  const float* gn_g2 = (const float*)d_in[8];
  const float* gn_be2 = (const float*)d_in[9];
  const float* sa_w3 = (const float*)d_in[10];
  const float* sa_b3 = (const float*)d_in[11];
  const float* ca_w1 = (const float*)d_in[12];
  const float* ca_b1 = (const float*)d_in[13];
  const float* ca_w2 = (const float*)d_in[14];
  const float* ca_b2 = (const float*)d_in[15];
  const float* conv_w = (const float*)d_in[16];
  float* out = (float*)d_out;

  float* ws = (float*)d_ws;
  float* stats1 = ws + 0;    // 16
  float* stats2 = ws + 16;   // 16
  float* sesum = ws + 32;    // 32
  float* f2 = ws + 64;       // 32
  float* t1 = ws + 128;                 // 2359296
  float* t2 = t1 + 2359296;             // 2359296
  float* Xc = t2 + 2359296;             // 4*589824
  float* P = Xc + 4 * (size_t)XSZ;      // 6*589824
  float* padb = P + 6 * (size_t)XSZ;    // 2560000

  hipMemsetAsync(d_ws, 0, 128 * sizeof(float), stream);

  dim3 blk(128);
  dim3 gConv(36, 1, BB);

  // --- spatial attention block ---
  k_conv1x1<<<gConv, blk, 0, stream>>>(x, sa_w1, sa_b1, t1);
  k_gnstats<<<dim3(72, BB), 256, 0, stream>>>(t1, stats1);
  k_gnapply<<<9216, 256, 0, stream>>>(t1, stats1, gn_g1, gn_be1);
  k_conv1x1<<<gConv, blk, 0, stream>>>(t1, sa_w2, sa_b2, t2);
  k_gnstats<<<dim3(72, BB), 256, 0, stream>>>(t2, stats2);
  k_gnapply<<<9216, 256, 0, stream>>>(t2, stats2, gn_g2, gn_be2);
  k_conv1x1<<<gConv, blk, 0, stream>>>(t2, sa_w3, sa_b3, t1);
  k_sigmul<<<1024, 256, 0, stream>>>(t1, x, sesum);

  // --- SE over relation dim ---
  k_se<<<1, 32, 0, stream>>>(sesum, ca_w1, ca_b1, ca_w2, ca_b2, f2);

  // --- FFT tensor product ---
  k_bflyx<<<2304, 256, 0, stream>>>(t1, f2, Xc);
  k_bigmm<<<dim3(4, 18, 6), blk, 0, stream>>>(adj, Xc, P);
  k_combine<<<2304, 256, 0, stream>>>(P, t2);

  // --- final 3x3 conv ---
  k_pad<<<10000, 256, 0, stream>>>(t2, padb);
  k_conv3<<<gConv, blk, 0, stream>>>(padb, conv_w, out);
}